// GCNGraphEncoder_23321672417918
// MI455X (gfx1250) — compile-verified
//
#include <hip/hip_runtime.h>

#define NNODES 100000
#define NEDGES 1250000
#define CH 64

typedef __attribute__((ext_vector_type(2))) float v2f;
typedef __attribute__((ext_vector_type(8))) float v8f;

// ---------------- degree / symmetric norm ----------------
__global__ void deg_init_kernel(float* __restrict__ deg) {
    int i = blockIdx.x * blockDim.x + threadIdx.x;
    if (i < NNODES) deg[i] = 1.0f;                 // self-loop contributes 1
}

__global__ void deg_edge_kernel(const int* __restrict__ dst, float* __restrict__ deg) {
    int e = blockIdx.x * blockDim.x + threadIdx.x;
    if (e < NEDGES) atomicAdd(&deg[dst[e]], 1.0f);
}

__global__ void dinv_kernel(float* __restrict__ deg) {
    int i = blockIdx.x * blockDim.x + threadIdx.x;
    if (i < NNODES) deg[i] = rsqrtf(deg[i]);       // deg >= 1 always (self-loops)
}

// ---------------- WMMA GEMM: Out[N x 64] = A[N x 64] @ W[64 x 64] (+ bias) ----
// 128 threads = 4 waves; block covers 64 rows x 64 cols.
// Each wave: 16 rows, four 16x16 tiles, K-loop of 16 x v_wmma_f32_16x16x4_f32.
#define XSTRIDE 66   // padded LDS stride for A slab (bank-conflict avoidance)

__global__ void __launch_bounds__(128)
gemm64_wmma_kernel(const float* __restrict__ A, const float* __restrict__ W,
                   const float* __restrict__ bias, float* __restrict__ Out,
                   int nRows) {
    __shared__ float sW[64 * 64];
    __shared__ float sA[64 * XSTRIDE];

    const int tid  = threadIdx.x;
    const int wave = tid >> 5;        // 0..3
    const int lane = tid & 31;
    const int half = lane >> 4;       // 0: lanes 0-15, 1: lanes 16-31
    const int lid  = lane & 15;
    const int rowBase = blockIdx.x * 64;

    // Stage W (64x64 = 16KB) into LDS, coalesced b128.
    {
        const float4* src = (const float4*)W;
        float4*       dst = (float4*)sW;
        for (int i = tid; i < 1024; i += 128) dst[i] = src[i];
    }
    // Stage 64-row x-slab into LDS (padded stride), zero-fill OOB rows.
    for (int i = tid; i < 64 * 16; i += 128) {
        int r = i >> 4, c4 = (i & 15) << 2;
        int gr = rowBase + r;
        float4 v = make_float4(0.f, 0.f, 0.f, 0.f);
        if (gr < nRows) v = *(const float4*)(A + (size_t)gr * CH + c4);
        float* d = &sA[r * XSTRIDE + c4];
        d[0] = v.x; d[1] = v.y; d[2] = v.z; d[3] = v.w;
    }
    __syncthreads();

    const int mRow = wave * 16;       // this wave's local row offset
    v8f acc[4] = {};                  // 4 column tiles: N0 = 0,16,32,48

    #pragma unroll
    for (int k0 = 0; k0 < 64; k0 += 4) {
        const int ka = k0 + 2 * half; // per-lane K index (same for A and B frags)
        v2f a;
        a.x = sA[(mRow + lid) * XSTRIDE + ka];
        a.y = sA[(mRow + lid) * XSTRIDE + ka + 1];
        #pragma unroll
        for (int nt = 0; nt < 4; ++nt) {
            v2f b;
            b.x = sW[ka * 64 + nt * 16 + lid];
            b.y = sW[(ka + 1) * 64 + nt * 16 + lid];
            acc[nt] = __builtin_amdgcn_wmma_f32_16x16x4_f32(
                false, a, false, b, (short)0, acc[nt], false, false);
        }
    }

    // C/D layout: VGPR r -> row (r + 8*half), col = lid (+ tile offset).
    #pragma unroll
    for (int nt = 0; nt < 4; ++nt) {
        #pragma unroll
        for (int r = 0; r < 8; ++r) {
            int gr = rowBase + mRow + r + 8 * half;
            int gc = nt * 16 + lid;
            if (gr < nRows) {
                float v = acc[nt][r];
                if (bias) v += bias[gc];
                Out[(size_t)gr * CH + gc] = v;
            }
        }
    }
}

// ---------------- aggregation ----------------
// agg[i,c] = h[i,c] * dinv[i]^2   (self-loop term; also initializes agg)
__global__ void selfloop_kernel(const float* __restrict__ h,
                                const float* __restrict__ dinv,
                                float* __restrict__ agg) {
    int idx = blockIdx.x * blockDim.x + threadIdx.x;
    if (idx < NNODES * CH) {
        float di = dinv[idx >> 6];
        agg[idx] = h[idx] * di * di;
    }
}

// agg[dst,c] += h[src,c] * dinv[src] * dinv[dst]  — one thread per (edge, channel);
// 64 consecutive channels -> contiguous 128B atomic bursts per wave.
__global__ void scatter_kernel(const float* __restrict__ h,
                               const float* __restrict__ dinv,
                               const int* __restrict__ src,
                               const int* __restrict__ dst,
                               float* __restrict__ agg) {
    int idx = blockIdx.x * blockDim.x + threadIdx.x;
    if (idx < NEDGES * CH) {
        int e = idx >> 6, c = idx & 63;
        int s = src[e], d = dst[e];
        float w = dinv[s] * dinv[d];
        atomicAdd(&agg[d * CH + c], h[s * CH + c] * w);
    }
}

__global__ void bias_relu_kernel(float* __restrict__ a, const float* __restrict__ b) {
    int idx = blockIdx.x * blockDim.x + threadIdx.x;
    if (idx < NNODES * CH) {
        float v = a[idx] + b[idx & 63];
        a[idx] = v > 0.f ? v : 0.f;
    }
}

// ---------------- driver ----------------
extern "C" void kernel_launch(void* const* d_in, const int* in_sizes, int n_in,
                              void* d_out, int out_size, void* d_ws, size_t ws_size,
                              hipStream_t stream) {
    const float* x  = (const float*)d_in[0];
    const int*   ei = (const int*)d_in[1];   // [2, E] flat: row0=src, row1=dst
    const float* W1 = (const float*)d_in[2];
    const float* b1 = (const float*)d_in[3];
    const float* W2 = (const float*)d_in[4];
    const float* b2 = (const float*)d_in[5];
    const float* W3 = (const float*)d_in[6];
    const float* b3 = (const float*)d_in[7];
    const int* src = ei;
    const int* dst = ei + NEDGES;

    // Workspace layout (floats): dinv[N] | h[N*64] | agg[N*64]  (~51.6 MB)
    float* ws   = (float*)d_ws;
    float* dinv = ws;
    float* h    = ws + 102400;
    float* agg  = h + (size_t)NNODES * CH;

    const int nB  = (NNODES + 255) / 256;
    const int eB  = (NEDGES + 255) / 256;
    const int ncB = (NNODES * CH + 255) / 256;
    const int ecB = (NEDGES * CH + 255) / 256;
    const int gG  = (NNODES + 63) / 64;

    deg_init_kernel<<<nB, 256, 0, stream>>>(dinv);
    deg_edge_kernel<<<eB, 256, 0, stream>>>(dst, dinv);
    dinv_kernel<<<nB, 256, 0, stream>>>(dinv);

    // Layer 1: GCNConv + ReLU
    gemm64_wmma_kernel<<<gG, 128, 0, stream>>>(x, W1, nullptr, h, NNODES);
    selfloop_kernel<<<ncB, 256, 0, stream>>>(h, dinv, agg);
    scatter_kernel<<<ecB, 256, 0, stream>>>(h, dinv, src, dst, agg);
    bias_relu_kernel<<<ncB, 256, 0, stream>>>(agg, b1);

    // Layer 2: GCNConv + ReLU
    gemm64_wmma_kernel<<<gG, 128, 0, stream>>>(agg, W2, nullptr, h, NNODES);
    selfloop_kernel<<<ncB, 256, 0, stream>>>(h, dinv, agg);
    scatter_kernel<<<ecB, 256, 0, stream>>>(h, dinv, src, dst, agg);
    bias_relu_kernel<<<ncB, 256, 0, stream>>>(agg, b2);

    // Layer 3: pure linear, bias fused into GEMM epilogue -> d_out
    gemm64_wmma_kernel<<<gG, 128, 0, stream>>>(agg, W3, b3, (float*)d_out, NNODES);
}